// NonLinearODELayer_78391743086675
// MI455X (gfx1250) — compile-verified
//
#include <hip/hip_runtime.h>

typedef __attribute__((ext_vector_type(2))) float v2f;
typedef __attribute__((ext_vector_type(4))) float v4f;
typedef __attribute__((ext_vector_type(8))) float v8f;

// out[b,k] = sum_{i,j} x[b,i] * W[k,i,j] * x[b,j]
// Symmetrized: out[b,k] = sum_d x_d^2 W[k,d,d] + sum_{i<j} x_i x_j (W[k,i,j]+W[k,j,i])
// -> GEMM [B x 136] * [136 x 16], K=136 swept as 34 x V_WMMA_F32_16X16X4_F32.
#define KC 136   // compacted K (16 diagonal + 120 upper-triangle products)
#define NG 34    // K-groups of 4

// c -> (i, j) map. Ordered so each group of 4 shares the A-row index i where
// possible (diag groups first, then fixed-i runs of 4, then 6 mixed groups).
constexpr int CI[KC] = {
    0, 1, 2, 3, 4, 5, 6, 7, 8, 9, 10, 11, 12, 13, 14, 15,          // diag
    0, 0, 0, 0, 0, 0, 0, 0, 0, 0, 0, 0,                            // (0, 1-12)
    1, 1, 1, 1, 1, 1, 1, 1, 1, 1, 1, 1,                            // (1, 2-13)
    2, 2, 2, 2, 2, 2, 2, 2, 2, 2, 2, 2,                            // (2, 3-14)
    3, 3, 3, 3, 3, 3, 3, 3, 3, 3, 3, 3,                            // (3, 4-15)
    4, 4, 4, 4, 4, 4, 4, 4,                                        // (4, 5-12)
    5, 5, 5, 5, 5, 5, 5, 5,                                        // (5, 6-13)
    6, 6, 6, 6, 6, 6, 6, 6,                                        // (6, 7-14)
    7, 7, 7, 7, 7, 7, 7, 7,                                        // (7, 8-15)
    8, 8, 8, 8,                                                    // (8, 9-12)
    9, 9, 9, 9,                                                    // (9, 10-13)
    10, 10, 10, 10,                                                // (10, 11-14)
    11, 11, 11, 11,                                                // (11, 12-15)
    0, 0, 0, 2,                                                    // mixed
    1, 1, 5, 5,
    4, 4, 4, 6,
    8, 8, 8, 10,
    9, 9, 13, 13,
    12, 12, 12, 14
};
constexpr int CJ[KC] = {
    0, 1, 2, 3, 4, 5, 6, 7, 8, 9, 10, 11, 12, 13, 14, 15,
    1, 2, 3, 4, 5, 6, 7, 8, 9, 10, 11, 12,
    2, 3, 4, 5, 6, 7, 8, 9, 10, 11, 12, 13,
    3, 4, 5, 6, 7, 8, 9, 10, 11, 12, 13, 14,
    4, 5, 6, 7, 8, 9, 10, 11, 12, 13, 14, 15,
    5, 6, 7, 8, 9, 10, 11, 12,
    6, 7, 8, 9, 10, 11, 12, 13,
    7, 8, 9, 10, 11, 12, 13, 14,
    8, 9, 10, 11, 12, 13, 14, 15,
    9, 10, 11, 12,
    10, 11, 12, 13,
    11, 12, 13, 14,
    12, 13, 14, 15,
    13, 14, 15, 15,
    14, 15, 14, 15,
    13, 14, 15, 15,
    13, 14, 15, 15,
    14, 15, 14, 15,
    13, 14, 15, 15
};

// Fold W [16,16,16] into compacted, symmetrized B matrix Vk [k][c] (k-major,
// stride KC) so the main kernel can preload lane-contiguous v2f fragments.
__global__ void build_Vk(const float* __restrict__ W, float* __restrict__ Vk) {
    int idx = blockIdx.x * blockDim.x + threadIdx.x;
    if (idx >= 16 * KC) return;
    int k = idx / KC;
    int c = idx - k * KC;
    int i = CI[c], j = CJ[c];
    float v = W[k * 256 + i * 16 + j];
    if (i != j) v += W[k * 256 + j * 16 + i];
    Vk[idx] = v;
}

// Half-wave operand select with *frontend* compile-time indices: collapses to
// a plain register read when LO == HI (no cndmask), else one value-level
// v_cndmask on the half-wave condition. Keeping indices template parameters
// guarantees SROA can promote xr[] fully to registers (round-2 regression was
// constexpr-table indices staying runtime loads at SROA time).
template <int LO, int HI_>
__device__ __forceinline__ float sel(const float (&xr)[16], bool hi) {
    if constexpr (LO == HI_) {
        return xr[LO];
    } else {
        return hi ? xr[HI_] : xr[LO];
    }
}

template <int G>
__device__ __forceinline__ void kstep(const float (&xr)[16], const v2f (&Bf)[NG],
                                      bool hi, v8f& acc) {
    constexpr int c0 = 4 * G;
    // Slot u: lanes 0-15 compute product c0+u, lanes 16-31 compute c0+2+u.
    float a0 = sel<CI[c0 + 0], CI[c0 + 2]>(xr, hi) *
               sel<CJ[c0 + 0], CJ[c0 + 2]>(xr, hi);
    float a1 = sel<CI[c0 + 1], CI[c0 + 3]>(xr, hi) *
               sel<CJ[c0 + 1], CJ[c0 + 3]>(xr, hi);
    v2f a;
    a.x = a0;
    a.y = a1;
    acc = __builtin_amdgcn_wmma_f32_16x16x4_f32(
        false, a, false, Bf[G], (short)0, acc, false, false);
}

template <int G>
__device__ __forceinline__ void ksweep(const float (&xr)[16], const v2f (&Bf)[NG],
                                       bool hi, v8f& acc) {
    kstep<G>(xr, Bf, hi, acc);
    if constexpr (G + 1 < NG) ksweep<G + 1>(xr, Bf, hi, acc);
}

__global__ __launch_bounds__(256) void bilinear_wmma_f32(
    const float* __restrict__ x,     // [B, 16]
    const float* __restrict__ Vk,    // [16, KC] symmetrized/compacted W
    float* __restrict__ out,         // [B, 16]
    int tiles)                       // B / 16
{
    const int lane = threadIdx.x & 31;
    const bool hi  = (lane >> 4) != 0;   // half-wave: K-slots {0,1} vs {2,3}
    const int n    = lane & 15;          // column k (B/D), row m (A)

    // Preload B fragments: 34 x v2f = 68 VGPRs, once per wave.
    // VGPR0 = K row 4g (lanes 0-15) / 4g+2 (lanes 16-31); VGPR1 = +1.
    v2f Bfrag[NG];
    {
        const float* vbase = Vk + n * KC + (hi ? 2 : 0);
        #pragma unroll
        for (int g = 0; g < NG; ++g)
            Bfrag[g] = *(const v2f*)(vbase + 4 * g);   // global_load_b64
    }

    const int wavesPerBlock = blockDim.x >> 5;
    const int waveId    = blockIdx.x * wavesPerBlock + (threadIdx.x >> 5);
    const int waveCount = gridDim.x * wavesPerBlock;

    for (int t = waveId; t < tiles; t += waveCount) {
        // Lane's batch row m = n: x[t*16 + m, :]
        const float* xrow = x + ((size_t)t * 16 + n) * 16;
        float xr[16];
        #pragma unroll
        for (int q = 0; q < 4; ++q) {
            v4f v = *(const v4f*)(xrow + 4 * q);       // global_load_b128
            xr[4 * q + 0] = v.x;
            xr[4 * q + 1] = v.y;
            xr[4 * q + 2] = v.z;
            xr[4 * q + 3] = v.w;
        }

        v8f acc = {};
        ksweep<0>(xr, Bfrag, hi, acc);                 // 34 x v_wmma_f32_16x16x4_f32

        // D: VGPR r, lane -> out[t*16 + r + 8*hi, n]
        float* obase = out + (size_t)t * 256 + (hi ? (size_t)128 : 0) + n;
        #pragma unroll
        for (int r = 0; r < 8; ++r)
            obase[r * 16] = acc[r];                    // global_store_b32
    }
}

extern "C" void kernel_launch(void* const* d_in, const int* in_sizes, int n_in,
                              void* d_out, int out_size, void* d_ws, size_t ws_size,
                              hipStream_t stream) {
    const float* x = (const float*)d_in[0];
    const float* W = (const float*)d_in[1];
    float* out = (float*)d_out;
    float* Vk = (float*)d_ws;       // 16*KC floats = 8704 bytes of scratch

    const int B = in_sizes[0] / 16;
    const int tiles = B / 16;

    // 1) Fold/symmetrize W -> Vk (tiny, one-shot, same stream).
    {
        const int total = 16 * KC;
        const int tpb = 256;
        hipLaunchKernelGGL(build_Vk, dim3((total + tpb - 1) / tpb), dim3(tpb),
                           0, stream, W, Vk);
    }

    // 2) Main GEMM-like sweep.
    {
        const int tpb = 256;                 // 8 wave32 per block
        const int wavesPerBlock = tpb / 32;
        int blocks = 2048;                   // 16384 waves -> 4 tiles/wave
        int maxBlocks = (tiles + wavesPerBlock - 1) / wavesPerBlock;
        if (blocks > maxBlocks) blocks = maxBlocks;
        if (blocks < 1) blocks = 1;
        hipLaunchKernelGGL(bilinear_wmma_f32, dim3(blocks), dim3(tpb), 0,
                           stream, x, Vk, out, tiles);
    }
}